// DescPairScanLoss_30992484008037
// MI455X (gfx1250) — compile-verified
//
#include <hip/hip_runtime.h>
#include <hip/hip_bf16.h>

typedef __attribute__((ext_vector_type(2))) float v2f;
typedef __attribute__((ext_vector_type(8))) float v8f;

#define BB 4
#define CC 128
#define MM 512

// ---------------------------------------------------------------------------
// Kernel 1: channel norms for anc/pos/neg -> ws.norms[3][B*M]
// ---------------------------------------------------------------------------
__global__ __launch_bounds__(256) void norms_kernel(const float* __restrict__ anc,
                                                    const float* __restrict__ pos,
                                                    const float* __restrict__ neg,
                                                    float* __restrict__ norms) {
    int t = blockIdx.x * blockDim.x + threadIdx.x;      // [0, 3*B*M)
    if (t >= 3 * BB * MM) return;
    int tensor = t >> 11;                               // / (B*M) = /2048
    int bm     = t & 2047;
    int b      = bm >> 9;                               // / M
    int m      = bm & 511;
    const float* p = (tensor == 0) ? anc : ((tensor == 1) ? pos : neg);
    const float* base = p + (size_t)b * CC * MM + m;
    float s = 0.0f;
#pragma unroll 4
    for (int c = 0; c < CC; ++c) {
        float v = base[(size_t)c * MM];
        s += v * v;
    }
    norms[t] = s;
}

// ---------------------------------------------------------------------------
// Kernel 2: per-batch weight normalizer M / sum(max(0, 3-sigma)); also zero
// the active-percentage accumulators at out[B*M + b].
// ---------------------------------------------------------------------------
__global__ __launch_bounds__(256) void wsum_kernel(const float* __restrict__ sigmas,
                                                   float* __restrict__ wscale,
                                                   float* __restrict__ out_active) {
    int b = blockIdx.x;
    __shared__ float red[256];
    float s = 0.0f;
    for (int m = threadIdx.x; m < MM; m += 256)
        s += fmaxf(3.0f - sigmas[b * MM + m], 0.0f);
    red[threadIdx.x] = s;
    __syncthreads();
    for (int off = 128; off > 0; off >>= 1) {
        if ((int)threadIdx.x < off) red[threadIdx.x] += red[threadIdx.x + off];
        __syncthreads();
    }
    if (threadIdx.x == 0) {
        wscale[b]     = (float)MM / red[0];
        out_active[b] = 0.0f;
    }
}

// ---------------------------------------------------------------------------
// Kernel 3: WMMA tile kernel. Grid = B * (M/16) row tiles, 8 waves per block.
// Waves 0..3 sweep pos column-tiles, waves 4..7 sweep neg column-tiles.
// dot-products via V_WMMA_F32_16X16X4_F32, running min(np - 2*dot) in regs.
// ---------------------------------------------------------------------------
__global__ __launch_bounds__(256) void tile_kernel(const float* __restrict__ anc,
                                                   const float* __restrict__ pos,
                                                   const float* __restrict__ neg,
                                                   const float* __restrict__ sigmas,
                                                   const float* __restrict__ norms,
                                                   const float* __restrict__ wscale,
                                                   float* __restrict__ out) {
    const int b    = blockIdx.x >> 5;        // batch
    const int rt   = blockIdx.x & 31;        // row tile (anchors)
    const int lane = threadIdx.x & 31;
    const int wave = threadIdx.x >> 5;
    const int hh   = lane >> 4;              // half: 0 -> rows 0..7, 1 -> rows 8..15
    const int l15  = lane & 15;
    const int koff = hh * 2;                 // f32 16x16x4 A/B layout k-offset

    // ---- preload A fragments (16 anchor columns x 128 k) into registers ----
    const int acol = rt * 16 + l15;
    const float* aptr = anc + (size_t)b * CC * MM + acol;
    v2f areg[32];
#pragma unroll
    for (int kk = 0; kk < 32; ++kk) {
        int k0 = kk * 4 + koff;
        v2f t;
        t.x = aptr[(size_t)k0 * MM];
        t.y = aptr[(size_t)(k0 + 1) * MM];
        areg[kk] = t;
    }

    const bool isNeg = (wave >= 4);
    const float* Bsrc = (isNeg ? neg : pos) + (size_t)b * CC * MM;
    const float* nb   = norms + (isNeg ? 2 * BB * MM : BB * MM) + b * MM;

    float rowmin[8];
#pragma unroll
    for (int r = 0; r < 8; ++r) rowmin[r] = 3.4e38f;

    // ---- sweep this wave's column tiles ----
    for (int ct = (wave & 3); ct < 32; ct += 4) {
        const int ncol = ct * 16 + l15;
        const float* bptr = Bsrc + ncol;
        v8f acc = {0.f, 0.f, 0.f, 0.f, 0.f, 0.f, 0.f, 0.f};
#pragma unroll
        for (int kk = 0; kk < 32; ++kk) {
            int k0 = kk * 4 + koff;
            v2f bf;
            bf.x = bptr[(size_t)k0 * MM];
            bf.y = bptr[(size_t)(k0 + 1) * MM];
            acc = __builtin_amdgcn_wmma_f32_16x16x4_f32(
                false, areg[kk], false, bf, (short)0, acc, false, false);
        }
        const float nbv = nb[ncol];
#pragma unroll
        for (int r = 0; r < 8; ++r)
            rowmin[r] = fminf(rowmin[r], nbv - 2.0f * acc[r]);
    }

    // ---- min across the 16 lanes of each half (wave32 xor shuffles) ----
#pragma unroll
    for (int msk = 8; msk >= 1; msk >>= 1) {
#pragma unroll
        for (int r = 0; r < 8; ++r)
            rowmin[r] = fminf(rowmin[r], __shfl_xor(rowmin[r], msk, 32));
    }

    __shared__ float smin[2][4][16];         // [pos/neg][wave slot][anchor row]
    if (l15 == 0) {
#pragma unroll
        for (int r = 0; r < 8; ++r)
            smin[(int)isNeg][wave & 3][hh * 8 + r] = rowmin[r];
    }
    __syncthreads();

    // ---- final combine: one wave, lanes 0..15 -> the 16 anchor rows ----
    if (wave == 0) {
        const int  row  = l15;
        const bool live = (lane < 16);
        float mp = 3.4e38f, mn = 3.4e38f;
#pragma unroll
        for (int w = 0; w < 4; ++w) {
            mp = fminf(mp, smin[0][w][row]);
            mn = fminf(mn, smin[1][w][row]);
        }
        const int gm = rt * 16 + row;
        const float na  = norms[b * MM + gm];
        const float dap = sqrtf(fmaxf(na + mp, 0.0f));
        const float dan = sqrtf(fmaxf(na + mn, 0.0f));
        const float bc  = dap - dan + 0.5f;
        const float wgt = fmaxf(3.0f - sigmas[b * MM + gm], 0.0f) * wscale[b];
        if (live) out[b * MM + gm] = wgt * fmaxf(bc, 0.0f);

        unsigned long long mask = __ballot(live && (bc > 0.0f));
        if (lane == 0) {
            int cnt = __popcll(mask & 0xFFFFull);
            atomicAdd(out + BB * MM + b, (float)cnt * (1.0f / (float)MM));
        }
    }
}

// ---------------------------------------------------------------------------
extern "C" void kernel_launch(void* const* d_in, const int* in_sizes, int n_in,
                              void* d_out, int out_size, void* d_ws, size_t ws_size,
                              hipStream_t stream) {
    const float* anc = (const float*)d_in[0];
    const float* pos = (const float*)d_in[1];
    const float* neg = (const float*)d_in[2];
    const float* sig = (const float*)d_in[3];
    float* out    = (float*)d_out;
    float* norms  = (float*)d_ws;                 // 3 * B*M floats
    float* wscale = norms + 3 * BB * MM;          // B floats

    norms_kernel<<<(3 * BB * MM + 255) / 256, 256, 0, stream>>>(anc, pos, neg, norms);
    wsum_kernel<<<BB, 256, 0, stream>>>(sig, wscale, out + BB * MM);
    tile_kernel<<<BB * (MM / 16), 256, 0, stream>>>(anc, pos, neg, sig, norms, wscale, out);
}